// FuzzyAttentionBase_75325136437360
// MI455X (gfx1250) — compile-verified
//
#include <hip/hip_runtime.h>
#include <math.h>
#include <float.h>

typedef __attribute__((ext_vector_type(2))) float v2f;
typedef __attribute__((ext_vector_type(4))) float v4f;
typedef __attribute__((ext_vector_type(8))) float v8f;

#define B_      4
#define T_      4096
#define R_      64
#define D_      128
#define TILE_T  16
#define NTILES  (T_ / TILE_T)     // 256
#define NBLOCKS (B_ * NTILES)     // 1024
#define EPS_F   1.17549435e-38f   // np.float32 tiny == FLT_MIN

// ---------------------------------------------------------------------------
// Prep kernel 1: B1[d][r] = s^2,  B2[b][d][r] = -2*k*s^2   (s = 1/(sigma+eps))
// ---------------------------------------------------------------------------
__global__ void prep_mats(const float* __restrict__ keys,
                          const float* __restrict__ sigma,
                          float* __restrict__ B1, float* __restrict__ B2) {
  int idx = blockIdx.x * blockDim.x + threadIdx.x;   // b*D*R + d*R + r
  if (idx >= B_ * D_ * R_) return;
  int b   = idx / (D_ * R_);
  int rem = idx % (D_ * R_);
  int d   = rem / R_;
  int r   = rem % R_;
  float s  = 1.0f / (sigma[r * D_ + d] + EPS_F);
  float s2 = s * s;
  if (b == 0) B1[d * R_ + r] = s2;
  B2[idx] = -2.0f * keys[((size_t)b * R_ + r) * D_ + d] * s2;
}

// ---------------------------------------------------------------------------
// Prep kernel 2: cterm[b][r] = sum_d (k*s)^2
// ---------------------------------------------------------------------------
__global__ void prep_cterm(const float* __restrict__ keys,
                           const float* __restrict__ sigma,
                           float* __restrict__ cterm) {
  int tid = threadIdx.x;            // 256 threads: b*64 + r
  int b = tid / R_, r = tid % R_;
  float acc = 0.f;
  for (int d = 0; d < D_; ++d) {
    float s  = 1.0f / (sigma[r * D_ + d] + EPS_F);
    float ks = keys[((size_t)b * R_ + r) * D_ + d] * s;
    acc += ks * ks;
  }
  cterm[tid] = acc;
}

// ---------------------------------------------------------------------------
// Main kernel: one block = 16 t-rows of one batch.
//   Phase A: fire logits via V_WMMA_F32_16X16X4_F32 (2 GEMM terms + c)
//   Phase B: per-row normalize, top-p threshold (rank-sums), renorm, compact
//   Phase C: importance partials + prediction over compacted survivors
// ---------------------------------------------------------------------------
__global__ __launch_bounds__(128) void fuzzy_main(
    const float* __restrict__ q, const float* __restrict__ values,
    const float* __restrict__ B1, const float* __restrict__ B2,
    const float* __restrict__ cterm, float* __restrict__ out,
    float* __restrict__ partials, float tau) {
  __shared__ float nfsLDS[TILE_T * R_];
  __shared__ float wLDS[TILE_T * R_];
  __shared__ int   listLDS[TILE_T * R_];
  __shared__ int   cntLDS[TILE_T];

  const int tid = threadIdx.x;
  const int blk = blockIdx.x;
  const int b   = blk / NTILES;
  const int t0  = (blk % NTILES) * TILE_T;

  // ---------------- Phase A: WMMA fire logits ----------------
  {
    const int lane  = tid & 31;
    const int wv    = tid >> 5;        // wave id = r-block (16 r's each)
    const int mrow  = lane & 15;       // A row / B col within tile
    const int hi    = lane >> 4;       // 0: K 0..1, 1: K 2..3
    const int khalf = hi << 1;
    const int rcol  = wv * 16 + mrow;

    const float* qrow = q + ((size_t)(b * T_ + t0 + mrow)) * D_;
    const float* B2b  = B2 + (size_t)b * D_ * R_;

    v8f acc = {0.f, 0.f, 0.f, 0.f, 0.f, 0.f, 0.f, 0.f};
    for (int k0 = 0; k0 < D_; k0 += 4) {
      float q0 = qrow[k0 + khalf];
      float q1 = qrow[k0 + khalf + 1];
      v2f a2 = {q0, q1};
      v2f a1 = {q0 * q0, q1 * q1};
      v2f b1 = {B1[(k0 + khalf) * R_ + rcol], B1[(k0 + khalf + 1) * R_ + rcol]};
      v2f b2 = {B2b[(k0 + khalf) * R_ + rcol], B2b[(k0 + khalf + 1) * R_ + rcol]};
      acc = __builtin_amdgcn_wmma_f32_16x16x4_f32(false, a1, false, b1,
                                                  (short)0, acc, false, false);
      acc = __builtin_amdgcn_wmma_f32_16x16x4_f32(false, a2, false, b2,
                                                  (short)0, acc, false, false);
    }
    const float cadd  = cterm[b * R_ + rcol];
    const float scale = -0.5f / (float)D_;
#pragma unroll
    for (int v = 0; v < 8; ++v) {
      int rowM = v + 8 * hi;           // C layout: vgpr v, lanes 16-31 -> M=v+8
      float fire = __expf(scale * (acc[v] + cadd)) + EPS_F;
      nfsLDS[rowM * R_ + rcol] = fire;
    }
  }
  __syncthreads();

  // ---------------- Phase B: per-row top-p ----------------
  {
    const int row = tid >> 3;          // 16 rows, 8 lanes each
    const int g   = tid & 7;           // lane within row group; r = g*8 + k
    float p[8];
    float tot = 0.f;
#pragma unroll
    for (int k = 0; k < 8; ++k) { p[k] = nfsLDS[row * R_ + g * 8 + k]; tot += p[k]; }
    tot += __shfl_xor(tot, 1, 8);
    tot += __shfl_xor(tot, 2, 8);
    tot += __shfl_xor(tot, 4, 8);
    float denom = tot + EPS_F;
#pragma unroll
    for (int k = 0; k < 8; ++k) p[k] = p[k] / denom;
#pragma unroll
    for (int k = 0; k < 8; ++k) nfsLDS[row * R_ + g * 8 + k] = p[k];
    __syncthreads();

    // rank sums: S_i = cumsum value at i's stable-descending-sort position
    float S[8] = {0.f, 0.f, 0.f, 0.f, 0.f, 0.f, 0.f, 0.f};
    for (int j = 0; j < R_; ++j) {
      float pj = nfsLDS[row * R_ + j];
#pragma unroll
      for (int k = 0; k < 8; ++k) {
        int ri = g * 8 + k;
        bool c = (pj > p[k]) || ((pj == p[k]) && (j <= ri));
        S[k] += c ? pj : 0.f;
      }
    }
    // first crossing: min S among S >= tau; threshold = that element's value
    float bestS = FLT_MAX, bestP = 0.f;
#pragma unroll
    for (int k = 0; k < 8; ++k)
      if (S[k] >= tau && S[k] < bestS) { bestS = S[k]; bestP = p[k]; }
#pragma unroll
    for (int m = 1; m < 8; m <<= 1) {
      float oS = __shfl_xor(bestS, m, 8);
      float oP = __shfl_xor(bestP, m, 8);
      if (oS < bestS) { bestS = oS; bestP = oP; }
    }
    float thresh = (bestS == FLT_MAX) ? 0.f : bestP;

    // mask, renormalize, deterministic compaction (r-ascending order)
    float qv[8];
    int   msk[8];
    float tot2 = 0.f;
    int   cntLoc = 0;
#pragma unroll
    for (int k = 0; k < 8; ++k) {
      int m = (p[k] >= thresh) ? 1 : 0;
      msk[k] = m;
      qv[k]  = m ? p[k] : 0.f;
      tot2  += qv[k];
      cntLoc += m;
    }
    tot2 += __shfl_xor(tot2, 1, 8);
    tot2 += __shfl_xor(tot2, 2, 8);
    tot2 += __shfl_xor(tot2, 4, 8);
    float denom2 = tot2 + EPS_F;

    int pre = cntLoc, u;
    u = __shfl_up(pre, 1, 8); if (g >= 1) pre += u;
    u = __shfl_up(pre, 2, 8); if (g >= 2) pre += u;
    u = __shfl_up(pre, 4, 8); if (g >= 4) pre += u;
    int excl = pre - cntLoc;
    int cntRow = __shfl(pre, 7, 8);
    if (g == 0) cntLDS[row] = cntRow;

    int idx = excl;
#pragma unroll
    for (int k = 0; k < 8; ++k) {
      float f = qv[k] / denom2;
      nfsLDS[row * R_ + g * 8 + k] = f;    // full (incl. zeros) for importance
      if (msk[k]) { listLDS[row * R_ + idx] = g * 8 + k; wLDS[row * R_ + idx] = f; idx++; }
    }
  }
  __syncthreads();

  // ---------------- Phase C1: importance partials (fixed order) -------------
  if (tid < R_) {
    float s = 0.f;
#pragma unroll
    for (int row = 0; row < TILE_T; ++row) s += nfsLDS[row * R_ + tid];
    partials[(size_t)blk * R_ + tid] = s;
  }

  // ---------------- Phase C2: prediction over survivors ---------------------
  {
    const int d = tid;                 // 128 threads == D
    float accs[TILE_T];
    const size_t vbase0 = ((size_t)(b * T_ + t0)) * R_ * D_ + d;
#pragma unroll 1
    for (int row = 0; row < TILE_T; ++row) {
      float a = 0.f;
      int cnt = cntLDS[row];
      const float* vb = values + vbase0 + (size_t)row * R_ * D_;
      for (int i = 0; i < cnt; ++i) {
        int   r = listLDS[row * R_ + i];
        float w = wLDS[row * R_ + i];
        a += w * vb[(size_t)r * D_];
      }
      accs[row] = a;
    }
    float* op = out + ((size_t)(b * D_ + d)) * T_ + t0;  // [B,D,T], 64B aligned
#pragma unroll
    for (int v = 0; v < 4; ++v) {
      v4f pk = {accs[4 * v], accs[4 * v + 1], accs[4 * v + 2], accs[4 * v + 3]};
      *(v4f*)(op + 4 * v) = pk;
    }
  }
}

// ---------------------------------------------------------------------------
// Loss kernel: importance = fixed-order sum of partials; var(ddof=1)/mean^2
// ---------------------------------------------------------------------------
__global__ void loss_kernel(const float* __restrict__ partials,
                            float* __restrict__ out) {
  __shared__ float imp[R_];
  int r = threadIdx.x;               // 64 threads
  float a = 0.f;
  for (int w = 0; w < NBLOCKS; ++w) a += partials[(size_t)w * R_ + r];
  imp[r] = a;
  __syncthreads();
  if (r == 0) {
    float s = 0.f;
    for (int i = 0; i < R_; ++i) s += imp[i];
    float mean = s / (float)R_;
    float v = 0.f;
    for (int i = 0; i < R_; ++i) { float dlt = imp[i] - mean; v += dlt * dlt; }
    v /= (float)(R_ - 1);
    out[(size_t)B_ * D_ * T_] = v / (mean * mean + 1e-10f);
  }
}

// ---------------------------------------------------------------------------
extern "C" void kernel_launch(void* const* d_in, const int* in_sizes, int n_in,
                              void* d_out, int out_size, void* d_ws, size_t ws_size,
                              hipStream_t stream) {
  (void)in_sizes; (void)n_in; (void)out_size; (void)ws_size;
  const float* queries = (const float*)d_in[0];  // [B,T,1,D]
  const float* keys    = (const float*)d_in[1];  // [B,1,R,D]
  const float* values  = (const float*)d_in[2];  // [B,T,R,D]
  const float* sigma   = (const float*)d_in[3];  // [R,D]
  float* out = (float*)d_out;                    // [B,D,T] ++ [loss]

  float* ws       = (float*)d_ws;
  float* B1       = ws;                          //  8192 f
  float* B2       = B1 + D_ * R_;                // 32768 f
  float* cterm    = B2 + B_ * D_ * R_;           //   256 f
  float* partials = cterm + B_ * R_;             // 65536 f  (total ~417 KB)

  // tau = 0.9 * (1/(R+eps))^(1/D), computed in double on host
  float tau = (float)(0.9 * pow(1.0 / (64.0 + 1.17549435e-38), 1.0 / 128.0));

  prep_mats<<<(B_ * D_ * R_ + 255) / 256, 256, 0, stream>>>(keys, sigma, B1, B2);
  prep_cterm<<<1, 256, 0, stream>>>(keys, sigma, cterm);
  fuzzy_main<<<NBLOCKS, 128, 0, stream>>>(queries, values, B1, B2, cterm, out,
                                          partials, tau);
  loss_kernel<<<1, R_, 0, stream>>>(partials, out);
}